// PinnMimics_6262062317735
// MI455X (gfx1250) — compile-verified
//
#include <hip/hip_runtime.h>
#include <math.h>

typedef __attribute__((ext_vector_type(2))) float v2f;
typedef __attribute__((ext_vector_type(8))) float v8f;

#define NSAMP 8192
#define NFEAT 8
#define HID   64
#define ROWS  128   // rows per block; 8 waves * 16-row stripes

// ---------------------------------------------------------------------------
// Branch-free fast transcendentals (v_exp_f32 / v_rcp_f32 / v_sqrt_f32).
// tanh saturates correctly at +-inf; ~1 ulp elsewhere.
// ---------------------------------------------------------------------------
__device__ __forceinline__ float fast_rcp(float x) { return __builtin_amdgcn_rcpf(x); }
__device__ __forceinline__ float fast_tanh(float x) {
  const float e = __expf(2.0f * x);
  return 1.0f - 2.0f * fast_rcp(e + 1.0f);
}
__device__ __forceinline__ float fast_sigm(float x) {
  return fast_rcp(1.0f + __expf(-x));
}
__device__ __forceinline__ float sigm_accurate(float x) {
  return 1.0f / (1.0f + expf(-x));
}

// ---------------------------------------------------------------------------
// Kernel 1: collapse the (sigma_orient) quadrature to 3 theta-moments and
// fold all scalar MIMICS parameters into 6 per-launch coefficients.
// Exact discrete identities over NPH=32 uniform phi samples:
//   sum cos = sum cos^3 = 0, sum cos^2 = 16, sum cos^4 = 12  (exact: NPH > 4).
// ---------------------------------------------------------------------------
__global__ void prep_kernel(const float* __restrict__ nb_raw,
                            const float* __restrict__ nl_raw,
                            const float* __restrict__ so_raw,
                            const float* __restrict__ mg_raw,
                            const float* __restrict__ s_raw,
                            float* __restrict__ ws) {
  __shared__ float sp[128], s22[128], s4c[128], s4s[128];
  const int i = threadIdx.x;
  const float PI_F = 3.14159265358979323846f;
  const float sig_o = (10.0f + 70.0f * sigm_accurate(so_raw[0])) * (PI_F / 180.0f);
  const float th = (float)i * (0.5f * PI_F) / 127.0f;  // linspace(0, pi/2, 128)
  const float d = th - 0.25f * PI_F;
  const float p = expf(-d * d / (2.0f * sig_o * sig_o)) * sinf(th);
  const float c = cosf(th), s = sinf(th);
  const float c2 = c * c, s2 = s * s;
  sp[i]  = p;
  s22[i] = p * c2 * s2;
  s4c[i] = p * c2 * c2;
  s4s[i] = p * s2 * s2;
  __syncthreads();
  if (i == 0) {
    float P = 0.f, A = 0.f, B = 0.f, C = 0.f;
    for (int k = 0; k < 128; ++k) { P += sp[k]; A += s22[k]; B += s4c[k]; C += s4s[k]; }
    const float S22 = A / P, S4c = B / P, S4s = C / P;
    const float LN10 = 2.302585092994046f;
    const float N_b = expf((2.0f + 3.0f * sigm_accurate(nb_raw[0])) * LN10);  // 10^x
    const float N_l = expf((3.0f + 3.0f * sigm_accurate(nl_raw[0])) * LN10);
    const float m_g = 0.05f + 0.75f * sigm_accurate(mg_raw[0]);
    const float s_m = 0.01f * (0.5f + 5.5f * sigm_accurate(s_raw[0]));
    const float eps_v = 1.5f + 20.0f * m_g;
    const float Kv = (eps_v - 1.0f) / (eps_v + 2.0f);
    const float Kv2 = Kv * Kv;
    const float dens = N_b * 1e-4f + N_l * 1e-6f;
    const float KW = (float)(2.0 * 3.14159265358979323846 * 5.405e9 / 2.998e8);
    float RC = 2.0f * KW * s_m; RC = RC * RC;            // rough = exp(-RC*ct^2)
    const float dk = dens * Kv2;
    ws[0] = dk * S4c;            // * ct^4
    ws[1] = 3.0f * dk * S22;     // * ct^2 st^2
    ws[2] = 0.375f * dk * S4s;   // * st^4
    ws[3] = 0.5f * dk * S22;     // vh * ct^2
    ws[4] = 0.125f * dk * S4s;   // vh * st^2
    ws[5] = RC;
  }
}

// ---------------------------------------------------------------------------
// One GEMM layer on the fp32 WMMA path (V_WMMA_F32_16X16X4_F32), fused
// branch-free tanh. Each wave computes one 16-row stripe x 4 column tiles.
// A fragment (16x4 f32): all lanes hold M = lane&15; v0/v1 = K pair
//   (lanes 0-15 -> K = k0+0/1, lanes 16-31 -> K = k0+2/3).
// B fragment (4x16 f32): lane holds N = lane&15, same K-pair split.
// C/D (16x16 f32, 8 VGPRs): row = i + 8*(lane>>4), col = lane&15.
// ---------------------------------------------------------------------------
template <int SA, int SB, int SO>
__device__ __forceinline__ void wmma_layer(const float (*Ain)[SA],
                                           const float (*Win)[SB],
                                           const float* __restrict__ bias,
                                           float (*Out)[SO],
                                           int K, int mt, int lane) {
  const int l16  = lane & 15;
  const int half = lane >> 4;
  const int kh   = half * 2;
  const int rowA = mt * 16 + l16;
  for (int nt = 0; nt < HID / 16; ++nt) {
    const int col = nt * 16 + l16;
    const float bv = bias[col];
    v8f c;
#pragma unroll
    for (int i = 0; i < 8; ++i) c[i] = bv;
    for (int k0 = 0; k0 < K; k0 += 4) {
      v2f a, b;
      a.x = Ain[rowA][k0 + kh];
      a.y = Ain[rowA][k0 + kh + 1];
      b.x = Win[k0 + kh][col];
      b.y = Win[k0 + kh + 1][col];
      c = __builtin_amdgcn_wmma_f32_16x16x4_f32(false, a, false, b,
                                                (short)0, c, false, false);
    }
#pragma unroll
    for (int i = 0; i < 8; ++i) {
      Out[mt * 16 + i + 8 * half][col] = fast_tanh(c[i]);
    }
  }
}

// ---------------------------------------------------------------------------
// Kernel 2: fused dual-MLP (WMMA f32) + closed-form MIMICS physics.
// Weights are staged LDS-direct via GLOBAL_LOAD_ASYNC_TO_LDS_B128 (ASYNCcnt).
// ---------------------------------------------------------------------------
__global__ __launch_bounds__(256) void pinn_mimics_kernel(
    const float* __restrict__ X, const float* __restrict__ theta,
    const float* __restrict__ Wp1, const float* __restrict__ bp1,
    const float* __restrict__ Wp2, const float* __restrict__ bp2,
    const float* __restrict__ Wp3, const float* __restrict__ bp3,
    const float* __restrict__ Wc1, const float* __restrict__ bc1,
    const float* __restrict__ Wc2, const float* __restrict__ bc2,
    const float* __restrict__ Wc3, const float* __restrict__ bc3,
    const float* __restrict__ coef, float* __restrict__ out) {
  __shared__ float sW1[NFEAT][HID];       // 16B-aligned rows (async B128 dest)
  __shared__ float sW2[HID][HID];         // unpadded: async B128 dest
  __shared__ float sW3[HID];
  __shared__ float sb1[HID];
  __shared__ float sb2[HID];
  __shared__ float sb3;
  __shared__ float sX[ROWS][NFEAT + 1];
  __shared__ float sH1[ROWS][HID + 1];    // stride 65: bank = (r+k) mod 64
  __shared__ float sH2[ROWS][HID + 1];
  __shared__ float sRes[2][ROWS];

  const int tid  = threadIdx.x;
  const int lane = tid & 31;
  const int wave = tid >> 5;
  const int row0 = blockIdx.x * ROWS;

  // Stage the 128x8 input tile (padded stride -> conflict-free A-frag loads).
  for (int idx = tid; idx < ROWS * NFEAT; idx += 256) {
    const int r = idx / NFEAT, c = idx % NFEAT;
    sX[r][c] = X[(row0 + r) * NFEAT + c];
  }

  const float* W1s[2] = {Wp1, Wc1};
  const float* B1s[2] = {bp1, bc1};
  const float* W2s[2] = {Wp2, Wc2};
  const float* B2s[2] = {bp2, bc2};
  const float* W3s[2] = {Wp3, Wc3};
  const float* B3s[2] = {bp3, bc3};

  for (int m = 0; m < 2; ++m) {
    __syncthreads();  // prior-iteration consumers done before async writes land

    // ---- async DMA: weights -> LDS (16B chunks, ASYNCcnt tracked) ----
    {
      const unsigned l2 = (unsigned)(uintptr_t)(&sW2[0][0]);
      const char* g2 = (const char*)W2s[m];
      for (int ch = tid; ch < (HID * HID) / 4; ch += 256) {   // 1024 chunks
        const unsigned laddr = l2 + (unsigned)(ch * 16);
        const char* ga = g2 + (size_t)ch * 16;
        asm volatile("global_load_async_to_lds_b128 %0, %1, off"
                     :: "v"(laddr), "v"(ga) : "memory");
      }
      const unsigned l1 = (unsigned)(uintptr_t)(&sW1[0][0]);
      const char* g1 = (const char*)W1s[m];
      if (tid < (NFEAT * HID) / 4) {                          // 128 chunks
        const unsigned laddr = l1 + (unsigned)(tid * 16);
        asm volatile("global_load_async_to_lds_b128 %0, %1, off"
                     :: "v"(laddr), "v"(g1 + (size_t)tid * 16) : "memory");
      }
    }
    if (tid < HID) {
      sW3[tid] = W3s[m][tid];
      sb1[tid] = B1s[m][tid];
      sb2[tid] = B2s[m][tid];
    }
    if (tid == 0) sb3 = B3s[m][0];
    asm volatile("s_wait_asynccnt 0x0" ::: "memory");
    __syncthreads();

    // Layer 1: (128x8)@(8x64), tanh.   2 WMMA K-steps per tile.
    wmma_layer(sX, sW1, sb1, sH1, NFEAT, wave, lane);
    __syncthreads();

    // Layer 2: (128x64)@(64x64), tanh. 16 WMMA K-steps per tile.
    wmma_layer(sH1, sW2, sb2, sH2, HID, wave, lane);
    __syncthreads();

    // Layer 3: (128x64)@(64x1) -> per-row scalar (VALU dot).
    if (tid < ROWS) {
      float acc = sb3;
#pragma unroll
      for (int k = 0; k < HID; ++k) acc = fmaf(sH2[tid][k], sW3[k], acc);
      sRes[m][tid] = acc;
    }
  }
  __syncthreads();

  // Fused elementwise physics (quadrature already collapsed to moments).
  if (tid < ROWS) {
    const int g = row0 + tid;
    const float m_v   = 0.93f * fast_sigm(sRes[0][tid]);
    const float delta = 0.05f * fast_tanh(sRes[1][tid]);
    const float mvf   = m_v + delta;
    const float eps_g = 3.0f + 25.0f * m_v + 10.0f * m_v * m_v;
    const float PI_F = 3.14159265358979323846f;
    const float ti = theta[g] * (PI_F / 180.0f);        // in [0.43, 0.96] rad
    const float ct = __cosf(ti), st = __sinf(ti);
    const float ct2 = ct * ct, st2 = st * st;
    const float Avv1 = coef[0], Avv2 = coef[1], Avv3 = coef[2];
    const float Avh1 = coef[3], Avh2 = coef[4], RC = coef[5];
    const float crown_vv = Avv1 * ct2 * ct2 + Avv2 * ct2 * st2 + Avv3 * st2 * st2;
    const float crown_vh = Avh1 * ct2 + Avh2 * st2;
    const float root = __builtin_amdgcn_sqrtf(eps_g - st2);
    const float rv = (eps_g * ct - root) * fast_rcp(eps_g * ct + root);
    const float ground = rv * rv * __expf(-RC * ct2) * ct2;
    const float svv = crown_vv + ground;
    const float svh = crown_vh + 0.05f * ground;
    const float L10 = 4.342944819032518f;  // 10 / ln(10)
    const float vv_db = L10 * __logf(svv + 1e-12f);
    const float vh_db = L10 * __logf(svh + 1e-12f);
    out[0 * NSAMP + g] = m_v;
    out[1 * NSAMP + g] = delta;
    out[2 * NSAMP + g] = mvf;
    out[3 * NSAMP + g] = vv_db;
    out[4 * NSAMP + g] = vh_db;
    out[5 * NSAMP + g] = eps_g;
  }
}

extern "C" void kernel_launch(void* const* d_in, const int* in_sizes, int n_in,
                              void* d_out, int out_size, void* d_ws, size_t ws_size,
                              hipStream_t stream) {
  const float* X   = (const float*)d_in[0];
  const float* th  = (const float*)d_in[1];
  // d_in[2] (vv_db_observed) is a pass-through diagnostic; not consumed.
  const float* Wp1 = (const float*)d_in[3];
  const float* bp1 = (const float*)d_in[4];
  const float* Wp2 = (const float*)d_in[5];
  const float* bp2 = (const float*)d_in[6];
  const float* Wp3 = (const float*)d_in[7];
  const float* bp3 = (const float*)d_in[8];
  const float* Wc1 = (const float*)d_in[9];
  const float* bc1 = (const float*)d_in[10];
  const float* Wc2 = (const float*)d_in[11];
  const float* bc2 = (const float*)d_in[12];
  const float* Wc3 = (const float*)d_in[13];
  const float* bc3 = (const float*)d_in[14];
  const float* nb  = (const float*)d_in[15];
  const float* nl  = (const float*)d_in[16];
  const float* so  = (const float*)d_in[17];
  const float* mg  = (const float*)d_in[18];
  const float* sr  = (const float*)d_in[19];
  float* ws = (float*)d_ws;

  prep_kernel<<<1, 128, 0, stream>>>(nb, nl, so, mg, sr, ws);
  pinn_mimics_kernel<<<NSAMP / ROWS, 256, 0, stream>>>(
      X, th, Wp1, bp1, Wp2, bp2, Wp3, bp3,
      Wc1, bc1, Wc2, bc2, Wc3, bc3, ws, (float*)d_out);
}